// HierarchicalGCN_20298015440982
// MI455X (gfx1250) — compile-verified
//
#include <hip/hip_runtime.h>

#define N_NODES   50000
#define N_EDGES   800000
#define N_GRAPHS  256
#define EMB       128
#define NLAYERS   8
#define N_CLASSES 10
#define BN_EPS    1e-5f
#define NEG_SLOPE 0.01f
#define BN_PARTS  512

typedef float v2f __attribute__((ext_vector_type(2)));
typedef float v8f __attribute__((ext_vector_type(8)));

// -------------------- edge scatter: agg[dst] += x[src] --------------------
__global__ __launch_bounds__(256)
void scatter_add_kernel(const float* __restrict__ x, const int* __restrict__ src,
                        const int* __restrict__ dst, float* __restrict__ agg, int nedges) {
    int e = blockIdx.x * 2 + (threadIdx.x >> 7);
    int c = threadIdx.x & 127;
    if (e < nedges) {
        int s = src[e], d = dst[e];
        atomicAdd(&agg[(size_t)d * EMB + c], x[(size_t)s * EMB + c]);
    }
}

// -------------------- GEMM: out = act((A [+Aagg]) @ W + bias) --------------------
// 256 threads = 8 waves; each wave computes a 16(M) x 128(N) strip with
// V_WMMA_F32_16X16X4_F32. W is staged in LDS in K-pair-interleaved form so each
// B fragment is a single aligned ds_load_b64. A path is branch-free (clamped
// pointer + 0/1 mask) and software-pipelined one K-step ahead of the WMMAs.
template<int ADD_AGG, int LEAKY>
__global__ __launch_bounds__(256)
void gemm128_kernel(const float* __restrict__ A, const float* __restrict__ Aagg,
                    const float* __restrict__ W, const float* __restrict__ bias,
                    float* __restrict__ out, int nrows) {
    // Wl[kp*256 + n*2 + j] = W[2*kp + j][n]  (kp = K-pair index 0..63)
    __shared__ float Wl[EMB * EMB];
    const int tid = threadIdx.x;
    {
        #pragma unroll
        for (int i = 0; i < 16; ++i) {
            int idx = tid + 256 * i;            // 0..4095, each covers 4 elements
            int kp  = idx >> 6;                 // 0..63
            int n0  = (idx & 63) * 2;           // 0,2,...,126
            float2 g0 = *(const float2*)(W + (size_t)(2 * kp) * EMB + n0);
            float2 g1 = *(const float2*)(W + (size_t)(2 * kp + 1) * EMB + n0);
            float4 wv = {g0.x, g1.x, g0.y, g1.y};
            *(float4*)(Wl + kp * 256 + n0 * 2) = wv;
        }
    }
    __syncthreads();

    const int wave = tid >> 5;
    const int lane = tid & 31;
    const int hsel = lane >> 4;        // which K-half of the fragment this lane holds
    const int r    = lane & 15;
    const int m0   = blockIdx.x * 128 + wave * 16;
    const int rowA = m0 + r;
    const bool  ok    = rowA < nrows;
    const float amask = ok ? 1.0f : 0.0f;
    const int   rowC  = ok ? rowA : (nrows - 1);
    const float* arow = A + (size_t)rowC * EMB + 2 * hsel;
    const float* grow = ADD_AGG ? (Aagg + (size_t)rowC * EMB + 2 * hsel) : nullptr;

    v8f acc[8];
    #pragma unroll
    for (int j = 0; j < 8; ++j)
        #pragma unroll
        for (int k = 0; k < 8; ++k) acc[j][k] = 0.0f;

    auto loadA = [&](int k) -> v2f {
        float2 a = *(const float2*)(arow + k);
        float ax = a.x, ay = a.y;
        if (ADD_AGG) {
            float2 g = *(const float2*)(grow + k);
            ax += g.x; ay += g.y;
        }
        v2f af;
        af.x = ax * amask;       // branch-free zeroing of out-of-range rows
        af.y = ay * amask;
        return af;
    };

    v2f a_cur = loadA(0);
    #pragma unroll
    for (int k = 0; k < EMB; k += 4) {
        v2f a_nxt;
        if (k + 4 < EMB) {
            a_nxt = loadA(k + 4);           // prefetch next fragment over the WMMAs
        } else {
            a_nxt.x = 0.0f; a_nxt.y = 0.0f;
        }
        const float* wp = Wl + (size_t)((k >> 1) + hsel) * 256 + r * 2;
        #pragma unroll
        for (int j = 0; j < 8; ++j) {
            v2f bf = *(const v2f*)(wp + j * 32);   // {W[kk][col], W[kk+1][col]}
            acc[j] = __builtin_amdgcn_wmma_f32_16x16x4_f32(
                false, a_cur, false, bf, (short)0, acc[j], false, false);
        }
        a_cur = a_nxt;
    }

    // epilogue: C/D layout -> VGPR e holds rows m0+e (lanes 0-15) / m0+8+e (lanes 16-31)
    #pragma unroll
    for (int j = 0; j < 8; ++j) {
        const int col = j * 16 + r;
        const float bcol = bias[col];
        #pragma unroll
        for (int e = 0; e < 8; ++e) {
            int row = m0 + e + 8 * hsel;
            if (row < nrows) {
                float v = acc[j][e] + bcol;
                if (LEAKY) v = (v >= 0.0f) ? v : NEG_SLOPE * v;
                out[(size_t)row * EMB + col] = v;
            }
        }
    }
}

// -------------------- BatchNorm stats (deterministic 2-stage) --------------------
__global__ __launch_bounds__(128)
void bn_partial_kernel(const float* __restrict__ h, float* __restrict__ psum,
                       float* __restrict__ psumsq, int nrows) {
    int c = threadIdx.x;
    float s = 0.0f, s2 = 0.0f;
    for (int r = blockIdx.x; r < nrows; r += gridDim.x) {
        float v = h[(size_t)r * EMB + c];
        s += v; s2 += v * v;
    }
    psum[blockIdx.x * EMB + c]   = s;
    psumsq[blockIdx.x * EMB + c] = s2;
}

__global__ __launch_bounds__(128)
void bn_reduce_kernel(const float* __restrict__ psum, const float* __restrict__ psumsq,
                      float* __restrict__ stats) {
    int c = threadIdx.x;
    float s = 0.0f, s2 = 0.0f;
    for (int b = 0; b < BN_PARTS; ++b) {
        s  += psum[b * EMB + c];
        s2 += psumsq[b * EMB + c];
    }
    float mu  = s * (1.0f / (float)N_NODES);
    float var = s2 * (1.0f / (float)N_NODES) - mu * mu;
    var = fmaxf(var, 0.0f);
    stats[c]       = mu;
    stats[EMB + c] = rsqrtf(var + BN_EPS);
}

// -------------------- BN apply + argmax (== hard gumbel-softmax) + outputs -----
// One wave per node; each lane handles 4 contiguous features.
__global__ __launch_bounds__(256)
void finalize_kernel(const float* __restrict__ h, const float* __restrict__ stats,
                     const float* __restrict__ gamma, const float* __restrict__ beta,
                     const float* __restrict__ gumbel, const int* __restrict__ batch,
                     float* __restrict__ nc, float* __restrict__ gc,
                     float* __restrict__ xout, int nrows) {
    int wave = threadIdx.x >> 5, lane = threadIdx.x & 31;
    int node = blockIdx.x * 8 + wave;
    if (node >= nrows) return;

    float4 hv = ((const float4*)(h      + (size_t)node * EMB))[lane];
    float4 gv = ((const float4*)(gumbel + (size_t)node * EMB))[lane];
    float4 mu = ((const float4*)(stats))[lane];
    float4 rs = ((const float4*)(stats + EMB))[lane];
    float4 ga = ((const float4*)(gamma))[lane];
    float4 be = ((const float4*)(beta))[lane];

    float ha[4] = {hv.x, hv.y, hv.z, hv.w};
    float gu[4] = {gv.x, gv.y, gv.z, gv.w};
    float mua[4] = {mu.x, mu.y, mu.z, mu.w};
    float rsa[4] = {rs.x, rs.y, rs.z, rs.w};
    float gaa[4] = {ga.x, ga.y, ga.z, ga.w};
    float bea[4] = {be.x, be.y, be.z, be.w};

    int c0 = lane * 4;
    float xo[4];
    float best = -3.0e38f;
    int bidx = 0;
    #pragma unroll
    for (int i = 0; i < 4; ++i) {
        float hn = (ha[i] - mua[i]) * rsa[i] * gaa[i] + bea[i];
        float z  = hn + gu[i];                 // tau=1; monotone -> argmax unchanged
        xo[i] = (hn >= 0.0f) ? hn : NEG_SLOPE * hn;
        if (z > best) { best = z; bidx = c0 + i; }   // first-index tie-break in-lane
    }
    // wave32 reduction, first-max-index tie-break
    #pragma unroll
    for (int off = 16; off > 0; off >>= 1) {
        float ob = __shfl_xor(best, off, 32);
        int   oi = __shfl_xor(bidx, off, 32);
        if (ob > best || (ob == best && oi < bidx)) { best = ob; bidx = oi; }
    }

    // node_concepts = exact one-hot (y_hard + y_soft - y_soft == y_hard)
    float4 ncv;
    ncv.x = (c0 + 0 == bidx) ? 1.0f : 0.0f;
    ncv.y = (c0 + 1 == bidx) ? 1.0f : 0.0f;
    ncv.z = (c0 + 2 == bidx) ? 1.0f : 0.0f;
    ncv.w = (c0 + 3 == bidx) ? 1.0f : 0.0f;
    ((float4*)(nc + (size_t)node * EMB))[lane] = ncv;

    float4 xov = {xo[0], xo[1], xo[2], xo[3]};
    ((float4*)(xout + (size_t)node * EMB))[lane] = xov;

    // segment_max of one-hots == set indicator (benign duplicate 1.0 stores)
    if (lane == 0) gc[(size_t)batch[node] * EMB + bidx] = 1.0f;
}

// -------------------- tiny classifier head --------------------
__global__ __launch_bounds__(32)
void logits_kernel(const float* __restrict__ gc, const float* __restrict__ W,
                   const float* __restrict__ b, float* __restrict__ out) {
    int g = blockIdx.x, c = threadIdx.x;
    if (c >= N_CLASSES) return;
    float s = b[c];
    #pragma unroll 8
    for (int k = 0; k < EMB; ++k) s += gc[(size_t)g * EMB + k] * W[k * N_CLASSES + c];
    out[(size_t)g * N_CLASSES + c] = s;
}

extern "C" void kernel_launch(void* const* d_in, const int* in_sizes, int n_in,
                              void* d_out, int out_size, void* d_ws, size_t ws_size,
                              hipStream_t stream) {
    const float* x_in   = (const float*)d_in[0];
    const int*   edge   = (const int*)d_in[1];   // [2, N_EDGES]: src then dst
    const int*   batch  = (const int*)d_in[2];
    const float* W1     = (const float*)d_in[3];
    const float* b1     = (const float*)d_in[4];
    const float* W2     = (const float*)d_in[5];
    const float* b2     = (const float*)d_in[6];
    const float* gamma  = (const float*)d_in[7];
    const float* beta   = (const float*)d_in[8];
    const float* gumbel = (const float*)d_in[9];
    const float* dW     = (const float*)d_in[10];
    const float* db     = (const float*)d_in[11];

    const size_t NF = (size_t)N_NODES * EMB;

    float* out     = (float*)d_out;
    float* logits  = out;                                   // 256*10
    float* nc_all  = out + (size_t)N_GRAPHS * N_CLASSES;    // 8*50000*128
    float* gc_all  = nc_all + (size_t)NLAYERS * NF;         // 8*256*128

    float* ws     = (float*)d_ws;
    float* agg    = ws;                 // NF
    float* mid    = agg + NF;           // NF
    float* hbuf   = mid + NF;           // NF
    float* xbuf   = hbuf + NF;          // NF
    float* psum   = xbuf + NF;          // BN_PARTS*128
    float* psumsq = psum + (size_t)BN_PARTS * EMB;
    float* stats  = psumsq + (size_t)BN_PARTS * EMB;        // 256

    // graph_concepts built by indicator stores -> zero it every call
    hipMemsetAsync(gc_all, 0, (size_t)NLAYERS * N_GRAPHS * EMB * sizeof(float), stream);

    const int gemm_blocks = (N_NODES + 127) / 128;
    const float* xcur = x_in;
    for (int i = 0; i < NLAYERS; ++i) {
        hipMemsetAsync(agg, 0, NF * sizeof(float), stream);
        scatter_add_kernel<<<(N_EDGES + 1) / 2, 256, 0, stream>>>(
            xcur, edge, edge + N_EDGES, agg, N_EDGES);
        gemm128_kernel<1, 1><<<gemm_blocks, 256, 0, stream>>>(
            xcur, agg, W1 + (size_t)i * EMB * EMB, b1 + i * EMB, mid, N_NODES);
        gemm128_kernel<0, 0><<<gemm_blocks, 256, 0, stream>>>(
            mid, nullptr, W2 + (size_t)i * EMB * EMB, b2 + i * EMB, hbuf, N_NODES);
        bn_partial_kernel<<<BN_PARTS, 128, 0, stream>>>(hbuf, psum, psumsq, N_NODES);
        bn_reduce_kernel<<<1, 128, 0, stream>>>(psum, psumsq, stats);
        finalize_kernel<<<(N_NODES + 7) / 8, 256, 0, stream>>>(
            hbuf, stats, gamma + i * EMB, beta + i * EMB,
            gumbel + (size_t)i * NF, batch,
            nc_all + (size_t)i * NF, gc_all + (size_t)i * N_GRAPHS * EMB,
            xbuf, N_NODES);
        xcur = xbuf;
    }
    logits_kernel<<<N_GRAPHS, 32, 0, stream>>>(
        gc_all + (size_t)(NLAYERS - 1) * N_GRAPHS * EMB, dW, db, logits);
}